// PhaseAdaptiveInput_54743653154900
// MI455X (gfx1250) — compile-verified
//
#include <hip/hip_runtime.h>
#include <stdint.h>

// ---------------------------------------------------------------------------
// PhaseAdaptiveInput: NNUE-style gather-accumulate.
//   out[b, :] = clip( sum_{f<32} W[idx[b,f], bucket(b), :] + bias[bucket(b), :],
//                     0, 1 )^2 * 255/256
// Pure gather-bandwidth workload (268 MB of 1KB random row reads,
// ~0.25 FLOP/byte -> HBM-bound, floor ~12us at 23.3 TB/s). WMMA does not
// apply (no shared dense operand). The MI455X-native path is the Tensor Data
// Mover in GATHER mode: one DMA pulls 16 indexed 1KB rows into LDS, tracked
// by TENSORcnt. Two TDM ops per row are software-pipelined against the LDS
// reduction via the split s_wait_tensorcnt counter.
// ---------------------------------------------------------------------------

#define SUM_OF_FEATURES 65536
#define OUTPUT_DIM      256
#define V4DIM           (OUTPUT_DIM / 4)   // 64 float4 lanes per row
#define BUCKET_SIZE     7                  // MAX_PLY(60) // COUNT(8)
#define NUM_FEATS       32
#define ROW_STRIDE_ELEM 2048               // COUNT * OUTPUT_DIM floats per W row
#define ACT_SCALE       (255.0f / 256.0f)

typedef __attribute__((ext_vector_type(4))) unsigned int v4u;
typedef __attribute__((ext_vector_type(8))) int          v8i;
typedef __attribute__((ext_vector_type(4))) int          v4i;

#ifndef __has_builtin
#define __has_builtin(x) 0
#endif

#if __has_builtin(__builtin_amdgcn_tensor_load_to_lds) && \
    __has_builtin(__builtin_amdgcn_s_wait_tensorcnt)
#define USE_TDM_GATHER 1
#else
#define USE_TDM_GATHER 0
#endif

__global__ __launch_bounds__(V4DIM)
void PhaseAdaptiveInput_kernel(const int*   __restrict__ feats,   // [B,32]
                               const int*   __restrict__ ply,     // [B]
                               const float* __restrict__ W,       // [65536,2048]
                               const float* __restrict__ bias,    // [2048]
                               float*       __restrict__ out)     // [B,256]
{
    __shared__ __align__(16) float tile[NUM_FEATS * OUTPUT_DIM]; // 32 KB

    const int row = blockIdx.x;
    const int t   = threadIdx.x;            // 0..63, owns dims [4t, 4t+3]

    // Uniform per block: bucket selection (scalar path).
    const int bucket = ply[row] / BUCKET_SIZE;
    const int* __restrict__ idx = feats + row * NUM_FEATS;

    float4 acc = make_float4(0.f, 0.f, 0.f, 0.f);

#if USE_TDM_GATHER
    // Wave 0 issues the two TDM gather DMAs (EXEC is ignored by tensor ops,
    // and the branch is wave-uniform under wave32).
    if (t < 32) {
        // ---- D# group 0: flags | lds_addr | global_addr | type ----
        const uint64_t gaddr =
            (uint64_t)(uintptr_t)W + (uint64_t)bucket * (OUTPUT_DIM * 4u);
        // Flat LDS addresses carry the LDS byte offset in the low 32 bits.
        const uint32_t lds_base = (uint32_t)(uintptr_t)(&tile[0]);

        v4u g0;
        g0.x = 0x80000001u;                       // count=1, gather_mode=1, 16b idx
        g0.y = lds_base;                          // per-op below
        g0.z = (uint32_t)gaddr;                   // global_addr[31:0]
        g0.w = ((uint32_t)(gaddr >> 32) & 0x01FFFFFFu) | (2u << 30); // [56:32]|type=2

        // ---- D# group 1: sizes / strides (gather mode) ----
        v8i g1;
        g1[0] = (int)(2u << 16);                  // wg_mask=0, data_size=4B
        g1[1] = (int)((uint32_t)OUTPUT_DIM << 16);// tensor_dim0[15:0]=256 @bits63:48
        g1[2] = 0;                                // tensor_dim0 hi, tensor_dim1 lo
        g1[3] = (int)(1u | ((uint32_t)OUTPUT_DIM << 16)); // dim1=65536 hi16, tile_dim0=256
        g1[4] = 16;                               // tile_dim1 = #valid indices
        g1[5] = ROW_STRIDE_ELEM;                  // tensor_dim0_stride = 2048 elems
        g1[6] = 0;
        g1[7] = 0;

        // ---- D# groups 2/3: packed 16-bit row indices ----
        v4i a2, a3, b2, b3;
        #pragma unroll
        for (int j = 0; j < 4; ++j) {
            a2[j] = (idx[     2*j] & 0xFFFF) | (idx[     2*j + 1] << 16);
            a3[j] = (idx[ 8 + 2*j] & 0xFFFF) | (idx[ 8 + 2*j + 1] << 16);
            b2[j] = (idx[16 + 2*j] & 0xFFFF) | (idx[16 + 2*j + 1] << 16);
            b3[j] = (idx[24 + 2*j] & 0xFFFF) | (idx[24 + 2*j + 1] << 16);
        }

#if defined(__clang_major__) && (__clang_major__ >= 23)
        // amdgpu-toolchain (clang-23 / therock-10.0): 6-arg builtin.
        v8i gx = {0, 0, 0, 0, 0, 0, 0, 0};
        v4u g0a = g0;                 // rows 0..15  -> tile[0 .. 16KB)
        __builtin_amdgcn_tensor_load_to_lds(g0a, g1, a2, a3, gx, 0);
        v4u g0b = g0;                 // rows 16..31 -> tile[16KB .. 32KB)
        g0b.y = lds_base + 16u * OUTPUT_DIM * 4u;
        __builtin_amdgcn_tensor_load_to_lds(g0b, g1, b2, b3, gx, 0);
#else
        // ROCm 7.2 (clang-22): 5-arg builtin.
        v4u g0a = g0;                 // rows 0..15  -> tile[0 .. 16KB)
        __builtin_amdgcn_tensor_load_to_lds(g0a, g1, a2, a3, 0);
        v4u g0b = g0;                 // rows 16..31 -> tile[16KB .. 32KB)
        g0b.y = lds_base + 16u * OUTPUT_DIM * 4u;
        __builtin_amdgcn_tensor_load_to_lds(g0b, g1, b2, b3, 0);
#endif
        // Tensor ops complete in order: cnt<=1 means the FIRST gather landed.
        __builtin_amdgcn_s_wait_tensorcnt(1);
    }
    __syncthreads();                              // rows 0..15 visible

    const float4* __restrict__ tv = (const float4*)tile;
    #pragma unroll
    for (int r = 0; r < 16; ++r) {                // reduce first half while
        float4 v = tv[r * V4DIM + t];             // second DMA is in flight
        acc.x += v.x; acc.y += v.y; acc.z += v.z; acc.w += v.w;
    }

    if (t < 32) __builtin_amdgcn_s_wait_tensorcnt(0);
    __syncthreads();                              // rows 16..31 visible

    #pragma unroll
    for (int r = 16; r < NUM_FEATS; ++r) {
        float4 v = tv[r * V4DIM + t];
        acc.x += v.x; acc.y += v.y; acc.z += v.z; acc.w += v.w;
    }
#else
    // Fallback: direct coalesced float4 gather-accumulate (no TDM builtin).
    const float4* __restrict__ Wv = (const float4*)W;
    #pragma unroll 4
    for (int r = 0; r < NUM_FEATS; ++r) {
        float4 v = Wv[(size_t)idx[r] * (ROW_STRIDE_ELEM / 4)
                      + bucket * V4DIM + t];
        acc.x += v.x; acc.y += v.y; acc.z += v.z; acc.w += v.w;
    }
#endif

    const float4 bv = ((const float4*)bias)[bucket * V4DIM + t];
    float4 x;
    x.x = fminf(fmaxf(acc.x + bv.x, 0.0f), 1.0f);
    x.y = fminf(fmaxf(acc.y + bv.y, 0.0f), 1.0f);
    x.z = fminf(fmaxf(acc.z + bv.z, 0.0f), 1.0f);
    x.w = fminf(fmaxf(acc.w + bv.w, 0.0f), 1.0f);

    float4 o;
    o.x = x.x * x.x * ACT_SCALE;
    o.y = x.y * x.y * ACT_SCALE;
    o.z = x.z * x.z * ACT_SCALE;
    o.w = x.w * x.w * ACT_SCALE;
    ((float4*)out)[(size_t)row * V4DIM + t] = o;
}

extern "C" void kernel_launch(void* const* d_in, const int* in_sizes, int n_in,
                              void* d_out, int out_size, void* d_ws, size_t ws_size,
                              hipStream_t stream) {
    const int*   feats = (const int*)  d_in[0];   // [B,32] int32
    const int*   ply   = (const int*)  d_in[1];   // [B]    int32
    const float* W     = (const float*)d_in[2];   // [65536, 2048] f32
    const float* bias  = (const float*)d_in[3];   // [2048] f32
    float*       out   = (float*)d_out;           // [B, 256] f32

    const int batch = in_sizes[1];                // ply element count == B
    PhaseAdaptiveInput_kernel<<<batch, V4DIM, 0, stream>>>(
        feats, ply, W, bias, out);
}